// TopKSparseInterconnect_1494648619382
// MI455X (gfx1250) — compile-verified
//
#include <hip/hip_runtime.h>
#include <cstdint>

#define BATCH   128
#define INPUTS  65536
#define OUTPUTS 65536
#define TOPK    32

// LDS tile row stride in floats: 32 + 4 pad. Keeps every float4 slot 16B-aligned
// (36*4 = 144 = 16*9) while breaking the worst bank-conflict pattern on the
// transposed read.
#define TSTRIDE 36

// Pointer-to-16B-vector types matching the async-LDS builtin's signature:
// clang declares __builtin_amdgcn_global_load_async_to_lds_b128 with
// (v4i addrspace(1)*, v4i addrspace(3)*, imm offset, imm cpol).
typedef int v4i __attribute__((vector_size(4 * sizeof(int))));
typedef __attribute__((address_space(1))) v4i gv4i;
typedef __attribute__((address_space(3))) v4i lv4i;

// ---------------------------------------------------------------------------
// Kernel 1: transpose x(BATCH x INPUTS) -> xT(INPUTS x BATCH) so that a
// gathered input column is 128 contiguous floats (512B, one b128/lane).
// Tiles are staged with CDNA5 async global->LDS copies (ASYNCcnt-tracked).
// ---------------------------------------------------------------------------
__global__ __launch_bounds__(256) void xpose_async_kernel(const float* __restrict__ x,
                                                          float* __restrict__ xT) {
  __shared__ float tile[32 * TSTRIDE];
  const int t  = threadIdx.x;
  const int i0 = blockIdx.x * 32;   // input-column tile base (2048 tiles)
  const int b0 = blockIdx.y * 32;   // batch tile base        (4 tiles)

  // Stage a 32x32 f32 tile: 256 lanes x 16B, fully coalesced along INPUTS.
  const int r  = t >> 3;            // 0..31 batch row within tile
  const int c4 = (t & 7) * 4;       // 0..28 input col within tile (float4)
  const float* gsrc = x + (size_t)(b0 + r) * INPUTS + (i0 + c4);
  float*       ldst = &tile[r * TSTRIDE + c4];

#if __has_builtin(__builtin_amdgcn_global_load_async_to_lds_b128)
  // CDNA5 async copy: memory -> LDS without a VGPR round trip.
  __builtin_amdgcn_global_load_async_to_lds_b128(
      (gv4i*)(uintptr_t)gsrc, (lv4i*)(uintptr_t)ldst, 0, 0);
#if __has_builtin(__builtin_amdgcn_s_wait_asynccnt)
  __builtin_amdgcn_s_wait_asynccnt(0);
#else
  asm volatile("s_wait_asynccnt 0" ::: "memory");
#endif
#else
  *(float4*)ldst = *(const float4*)gsrc;          // fallback (host pass etc.)
#endif
  __syncthreads();

  // Read transposed and store: each thread emits one float4 row-segment of xT.
  const int i  = t >> 3;            // 0..31 input col  (row of xT)
  const int b4 = (t & 7) * 4;       // 0..28 batch within tile
  float4 v;
  v.x = tile[(b4 + 0) * TSTRIDE + i];
  v.y = tile[(b4 + 1) * TSTRIDE + i];
  v.z = tile[(b4 + 2) * TSTRIDE + i];
  v.w = tile[(b4 + 3) * TSTRIDE + i];
  *(float4*)(xT + (size_t)(i0 + i) * BATCH + (b0 + b4)) = v;
}

// ---------------------------------------------------------------------------
// Kernel 2: one output column per wave32.
//   Phase 1 (lane == k):    load idx/c, softmax via xor-shuffle reductions.
//   Phase 2 (lane == b/4):  v_readlane puts (idx_k, w_k) into SGPRs; each of
//                           the 32 unrolled steps is one coalesced 512B b128
//                           load of an xT column (L2-resident) + 4 scalar-
//                           operand FMACs.
// ---------------------------------------------------------------------------
__global__ __launch_bounds__(256) void topk_wsum_kernel(const float* __restrict__ xT,
                                                        const float* __restrict__ top_c,
                                                        const int*   __restrict__ top_idx,
                                                        float* __restrict__ out) {
  const int lane = threadIdx.x & 31;
  const int o    = blockIdx.x * 8 + (threadIdx.x >> 5);   // 8 waves per block

  // ---- phase 1: lanes are the TOPK axis ----
  int   my_idx = top_idx[(size_t)lane * OUTPUTS + o];
  float my_c   = top_c  [(size_t)lane * OUTPUTS + o];

  float m = my_c;
#pragma unroll
  for (int off = 16; off; off >>= 1) m = fmaxf(m, __shfl_xor(m, off, 32));
  const float e = __expf(my_c - m);
  float s = e;
#pragma unroll
  for (int off = 16; off; off >>= 1) s += __shfl_xor(s, off, 32);
  const int my_w = __float_as_int(e / s);   // lane k holds softmax weight w_k

  // ---- phase 2: lanes are the batch axis (4 batches per lane) ----
  float4 acc = make_float4(0.f, 0.f, 0.f, 0.f);
#pragma unroll
  for (int k = 0; k < TOPK; ++k) {
    const int   sidx = __builtin_amdgcn_readlane(my_idx, k);                  // -> SGPR
    const float sw   = __int_as_float(__builtin_amdgcn_readlane(my_w, k));    // -> SGPR
    const float4 v = *(const float4*)(xT + (size_t)sidx * BATCH + lane * 4);  // 512B/wave
    acc.x = fmaf(sw, v.x, acc.x);
    acc.y = fmaf(sw, v.y, acc.y);
    acc.z = fmaf(sw, v.z, acc.z);
    acc.w = fmaf(sw, v.w, acc.w);
  }

  const int b = lane * 4;
  out[(size_t)(b + 0) * OUTPUTS + o] = acc.x;   // column-scattered stores:
  out[(size_t)(b + 1) * OUTPUTS + o] = acc.y;   // absorbed/combined in 192MB L2
  out[(size_t)(b + 2) * OUTPUTS + o] = acc.z;
  out[(size_t)(b + 3) * OUTPUTS + o] = acc.w;
}

// ---------------------------------------------------------------------------
extern "C" void kernel_launch(void* const* d_in, const int* in_sizes, int n_in,
                              void* d_out, int out_size, void* d_ws, size_t ws_size,
                              hipStream_t stream) {
  const float* x       = (const float*)d_in[0];   // (128, 65536) f32
  const float* top_c   = (const float*)d_in[1];   // (32, 65536)  f32
  const int*   top_idx = (const int*)  d_in[2];   // (32, 65536)  int
  float*       out     = (float*)d_out;           // (128, 65536) f32
  float*       xT      = (float*)d_ws;            // needs 65536*128*4 = 32 MB scratch

  // 1) x -> xT transpose (async-LDS staged), fully rewritten every call.
  xpose_async_kernel<<<dim3(INPUTS / 32, BATCH / 32), 256, 0, stream>>>(x, xT);

  // 2) gather + softmax-weighted reduction: 65536 columns, 1 column per wave,
  //    8 waves (256 threads) per block.
  topk_wsum_kernel<<<OUTPUTS / 8, 256, 0, stream>>>(xT, top_c, top_idx, out);
}